// OurGCN_75273596830285
// MI455X (gfx1250) — compile-verified
//
#include <hip/hip_runtime.h>
#include <math.h>

typedef float v2f __attribute__((ext_vector_type(2)));
typedef float v8f __attribute__((ext_vector_type(8)));

#define NNODES 100000
#define NEDGES 1600000
#define FIN    256
#define FH     64
#define FC     40

// ---------------- gcn_norm kernels ----------------

__global__ void k_deg_init(float* __restrict__ deg) {
    int i = blockIdx.x * blockDim.x + threadIdx.x;
    if (i < NNODES) deg[i] = 1.0f;   // self-loop weight contributes 1 to degree
}

__global__ void k_deg_acc(const int* __restrict__ col, const float* __restrict__ w,
                          float* __restrict__ deg) {
    int e = blockIdx.x * blockDim.x + threadIdx.x;
    if (e < NEDGES) atomicAdd(&deg[col[e]], w[e]);
}

__global__ void k_dinv(const float* __restrict__ deg, float* __restrict__ dis) {
    int i = blockIdx.x * blockDim.x + threadIdx.x;
    if (i < NNODES) {
        float d = deg[i];
        dis[i] = (d > 0.0f) ? rsqrtf(d) : 0.0f;
    }
}

__global__ void k_norm(const int* __restrict__ row, const int* __restrict__ col,
                       const float* __restrict__ w, const float* __restrict__ dis,
                       float* __restrict__ nrm) {
    int e = blockIdx.x * blockDim.x + threadIdx.x;
    if (e < NEDGES) nrm[e] = dis[row[e]] * w[e] * dis[col[e]];
}

// ---------------- fp32 WMMA GEMM:  C[M x NC] = A[M x K] * B[K x NC] ----------------
// One wave per 16-row tile; wave accumulates all NCPAD/16 column tiles so the A
// fragment is loaded once per k-step (A traffic = M*K*4 exactly) and the WMMA
// chains are independent.  B is staged through LDS in 64-row K-slabs using a
// K-pair-interleaved layout: element (k,n) at [(k>>1)*S + 2n + (k&1)], so each
// B fragment is ONE aligned ds_load_b64 straight into an even VGPR pair (no
// repacking movs).  S is chosen == 32 (mod 64) words so the two half-waves
// (which differ by one S row) map to disjoint bank halves -> conflict-free.

template <int K, int NC, int NCPAD>
__global__ __launch_bounds__(256) void k_gemm_wmma(const float* __restrict__ A,
                                                   const float* __restrict__ B,
                                                   float* __restrict__ C, int M) {
    constexpr int COLT = NCPAD / 16;
    // smallest S >= 2*NCPAD with S % 64 == 32  (words)
    constexpr int S = ((2 * NCPAD - 32 + 63) / 64) * 64 + 32;
    __shared__ float Blds[32 * S];             // 32 K-pair rows per 64-K slab

    const int lane  = threadIdx.x & 31;
    const int wave  = threadIdx.x >> 5;
    const int tileM = blockIdx.x * 8 + wave;
    const int rowTiles = M >> 4;
    const bool valid = (tileM < rowTiles);     // wave-uniform

    const int half = lane >> 4;                // 0: K pair {0,1}; 1: K pair {2,3}
    const int m    = lane & 15;                // A: row-in-tile, B: col-in-tile

    v8f acc[COLT] = {};
    const float* Arow = A + (size_t)(tileM * 16 + m) * K;

    for (int ks = 0; ks < K; ks += 64) {
        __syncthreads();
        // cooperative slab load of B[ks..ks+64) x NCPAD, pair-interleaved
        for (int i = threadIdx.x; i < 64 * NCPAD; i += 256) {
            int kk = i / NCPAD, nn = i % NCPAD;
            float v = (nn < NC) ? B[(size_t)(ks + kk) * NC + nn] : 0.0f;
            Blds[(kk >> 1) * S + 2 * nn + (kk & 1)] = v;
        }
        __syncthreads();

        if (valid) {   // uniform per wave -> EXEC all ones inside (WMMA requirement)
#pragma unroll
            for (int k0 = 0; k0 < 64; k0 += 4) {
                // A fragment (16x4 f32): lane m, K pair {k0+2h, k0+2h+1} -> one b64
                v2f a;
                const float* ap = Arow + ks + k0 + 2 * half;
                a.x = ap[0];
                a.y = ap[1];
                const int kp = (k0 >> 1) + half;           // K-pair row in LDS
                const v2f* bp = (const v2f*)&Blds[kp * S + 2 * m];
#pragma unroll
                for (int t = 0; t < COLT; t++) {
                    v2f b = bp[t * 16];                    // single ds_load_b64
                    acc[t] = __builtin_amdgcn_wmma_f32_16x16x4_f32(
                        false, a, false, b, (short)0, acc[t], false, false);
                }
            }
        }
    }

    if (valid) {
#pragma unroll
        for (int t = 0; t < COLT; t++) {
            int coln = t * 16 + m;
            if (coln < NC) {
#pragma unroll
                for (int r = 0; r < 8; r++) {
                    int rown = tileM * 16 + r + 8 * half;   // C/D VGPR layout
                    C[(size_t)rown * NC + coln] = acc[t][r];
                }
            }
        }
    }
}

// ---------------- aggregation (edge-parallel, L2-resident f32 atomics) ----------------

// out[i,:] = dis[i]^2 * H[i,:]   (self-loop term; also zero-initializes accumulator)
__global__ void k_selfscale(const float* __restrict__ H, const float* __restrict__ dis,
                            float* __restrict__ Aout, int NC) {
    int idx = blockIdx.x * blockDim.x + threadIdx.x;
    if (idx < NNODES * NC) {
        int i = idx / NC;
        float s = dis[i];
        Aout[idx] = s * s * H[idx];
    }
}

// layer-1 messages: 16 threads/edge, float4 per thread (64 features)
__global__ void k_agg_edges_h(const int* __restrict__ row, const int* __restrict__ col,
                              const float* __restrict__ nrm, const float* __restrict__ H,
                              float* __restrict__ Aout) {
    int t = blockIdx.x * blockDim.x + threadIdx.x;
    int e = t >> 4, seg = t & 15;
    if (e < NEDGES) {
        int r = row[e], c = col[e];
        float nv = nrm[e];
        const float4 v = ((const float4*)(H + (size_t)r * FH))[seg];
        float* dst = Aout + (size_t)c * FH + seg * 4;
        atomicAdd(dst + 0, nv * v.x);
        atomicAdd(dst + 1, nv * v.y);
        atomicAdd(dst + 2, nv * v.z);
        atomicAdd(dst + 3, nv * v.w);
    }
}

// layer-2 messages: 8 threads/edge, 5 floats per thread (40 features)
__global__ void k_agg_edges_c(const int* __restrict__ row, const int* __restrict__ col,
                              const float* __restrict__ nrm, const float* __restrict__ Z,
                              float* __restrict__ F) {
    int t = blockIdx.x * blockDim.x + threadIdx.x;
    int e = t >> 3, seg = t & 7;
    if (e < NEDGES) {
        int r = row[e], c = col[e];
        float nv = nrm[e];
        const float* src = Z + (size_t)r * FC + seg * 5;
        float* dst = F + (size_t)c * FC + seg * 5;
#pragma unroll
        for (int j = 0; j < 5; j++) atomicAdd(dst + j, nv * src[j]);
    }
}

__global__ void k_relu_bias(float* __restrict__ A, const float* __restrict__ b) {
    int idx = blockIdx.x * blockDim.x + threadIdx.x;
    if (idx < NNODES * FH) {
        float v = A[idx] + b[idx & (FH - 1)];
        A[idx] = v > 0.0f ? v : 0.0f;
    }
}

// final = F + b2 ; out2 = log_softmax(final)  (both written to d_out)
__global__ void k_final(const float* __restrict__ F, const float* __restrict__ b2,
                        float* __restrict__ out) {
    int i = blockIdx.x * blockDim.x + threadIdx.x;
    if (i >= NNODES) return;
    float v[FC];
    float mx = -INFINITY;
    const float* src = F + (size_t)i * FC;
#pragma unroll
    for (int c = 0; c < FC; c++) { v[c] = src[c] + b2[c]; mx = fmaxf(mx, v[c]); }
    float s = 0.0f;
#pragma unroll
    for (int c = 0; c < FC; c++) s += expf(v[c] - mx);
    float lse = mx + logf(s);
    float* o1 = out + (size_t)i * FC;
    float* o2 = out + (size_t)NNODES * FC + (size_t)i * FC;
#pragma unroll
    for (int c = 0; c < FC; c++) { o1[c] = v[c]; o2[c] = v[c] - lse; }
}

// ---------------- launcher ----------------

extern "C" void kernel_launch(void* const* d_in, const int* in_sizes, int n_in,
                              void* d_out, int out_size, void* d_ws, size_t ws_size,
                              hipStream_t stream) {
    const float* x  = (const float*)d_in[0];
    const int*   ei = (const int*)d_in[1];     // [2, E] int32
    const float* ew = (const float*)d_in[2];
    const float* W1 = (const float*)d_in[3];
    const float* b1 = (const float*)d_in[4];
    const float* W2 = (const float*)d_in[5];
    const float* b2 = (const float*)d_in[6];
    float* out = (float*)d_out;

    const int* row = ei;
    const int* col = ei + NEDGES;

    float* ws  = (float*)d_ws;
    float* deg = ws;  ws += NNODES;
    float* dis = ws;  ws += NNODES;
    float* nrm = ws;  ws += NEDGES;
    float* H0  = ws;  ws += (size_t)NNODES * FH;
    float* A1  = ws;  ws += (size_t)NNODES * FH;   // aggregated; ReLU done in place
    float* Z   = ws;  ws += (size_t)NNODES * FC;
    float* F   = ws;  // NNODES * FC

    dim3 blk(256);

    // gcn_norm
    k_deg_init<<<(NNODES + 255) / 256, blk, 0, stream>>>(deg);
    k_deg_acc <<<(NEDGES + 255) / 256, blk, 0, stream>>>(col, ew, deg);
    k_dinv    <<<(NNODES + 255) / 256, blk, 0, stream>>>(deg, dis);
    k_norm    <<<(NEDGES + 255) / 256, blk, 0, stream>>>(row, col, ew, dis, nrm);

    // layer 1: H0 = x @ W1 (fp32 WMMA), aggregate, ReLU(+b1)
    k_gemm_wmma<FIN, FH, 64><<<(NNODES / 16 + 7) / 8, blk, 0, stream>>>(x, W1, H0, NNODES);
    k_selfscale  <<<((size_t)NNODES * FH + 255) / 256, blk, 0, stream>>>(H0, dis, A1, FH);
    k_agg_edges_h<<<((size_t)NEDGES * 16 + 255) / 256, blk, 0, stream>>>(row, col, nrm, H0, A1);
    k_relu_bias  <<<((size_t)NNODES * FH + 255) / 256, blk, 0, stream>>>(A1, b1);

    // layer 2: Z = h @ W2 (fp32 WMMA, NC=40 padded to 48), aggregate
    k_gemm_wmma<FH, FC, 48><<<(NNODES / 16 + 7) / 8, blk, 0, stream>>>(A1, W2, Z, NNODES);
    k_selfscale  <<<((size_t)NNODES * FC + 255) / 256, blk, 0, stream>>>(Z, dis, F, FC);
    k_agg_edges_c<<<((size_t)NEDGES * 8 + 255) / 256, blk, 0, stream>>>(row, col, nrm, Z, F);

    // bias + log_softmax, write both outputs
    k_final<<<(NNODES + 255) / 256, blk, 0, stream>>>(F, b2, out);
}